// EfficientEmbeddingLayer_37864431681677
// MI455X (gfx1250) — compile-verified
//
#include <hip/hip_runtime.h>
#include <math.h>

#define VOCAB  50257
#define HIDDEN 1024
#define BATCH  4
#define SEQ    512
#define EPS    1e-5f

typedef float v4f __attribute__((ext_vector_type(4)));

// ---- ordered-uint encoding of float so we can atomicMin/Max on uint ----
__device__ __forceinline__ unsigned orderKey(float f) {
    unsigned u = __float_as_uint(f);
    return (u & 0x80000000u) ? ~u : (u | 0x80000000u);
}
__device__ __forceinline__ float keyToFloat(unsigned k) {
    unsigned u = (k & 0x80000000u) ? (k ^ 0x80000000u) : ~k;
    return __uint_as_float(u);
}

// ws[0] = encoded running min, ws[1] = encoded running max.
// ws is poisoned by the harness; re-seed every launch (deterministic).
__global__ void init_ws_kernel(unsigned* __restrict__ ws) {
    ws[0] = 0xFFFFFFFFu;  // max possible key -> any real value lowers it
    ws[1] = 0x00000000u;  // min possible key -> any real value raises it
}

// Pass 1: streaming min/max over the 206MB weight matrix.
// NT loads: stream is larger than the 192MB L2; don't thrash the cache the
// gather pass wants to hit.
__global__ __launch_bounds__(256)
void minmax_kernel(const float* __restrict__ w, unsigned* __restrict__ ws, int n4) {
    const v4f* __restrict__ w4 = (const v4f*)w;
    float lmin = __uint_as_float(0x7F800000u);   // +inf
    float lmax = __uint_as_float(0xFF800000u);   // -inf
    const int stride = gridDim.x * blockDim.x;
    for (int i = blockIdx.x * blockDim.x + threadIdx.x; i < n4; i += stride) {
        v4f v = __builtin_nontemporal_load(&w4[i]);
        lmin = fminf(lmin, fminf(fminf(v.x, v.y), fminf(v.z, v.w)));
        lmax = fmaxf(lmax, fmaxf(fmaxf(v.x, v.y), fmaxf(v.z, v.w)));
    }
    // wave32 butterfly reduction (CDNA5 is wave32-only)
    #pragma unroll
    for (int off = 16; off > 0; off >>= 1) {
        lmin = fminf(lmin, __shfl_xor(lmin, off, 32));
        lmax = fmaxf(lmax, __shfl_xor(lmax, off, 32));
    }
    __shared__ float smin[8], smax[8];   // 256 threads = 8 waves
    const int wave = threadIdx.x >> 5;
    if ((threadIdx.x & 31) == 0) { smin[wave] = lmin; smax[wave] = lmax; }
    __syncthreads();
    if (threadIdx.x == 0) {
        float m = smin[0], M = smax[0];
        #pragma unroll
        for (int i = 1; i < 8; ++i) { m = fminf(m, smin[i]); M = fmaxf(M, smax[i]); }
        atomicMin(&ws[0], orderKey(m));
        atomicMax(&ws[1], orderKey(M));
    }
}

// Pass 2: one block per (b,s) token row. 256 threads x 4 floats = 1024 hidden.
// Gather the embedding row via CDNA5 async global->LDS b128 copies, dequantize,
// add on-the-fly sinusoidal PE, then LayerNorm the row.
__global__ __launch_bounds__(256)
void embed_ln_kernel(const int*   __restrict__ ids,
                     const float* __restrict__ weight,
                     const float* __restrict__ gamma,
                     const float* __restrict__ beta,
                     const unsigned* __restrict__ ws,
                     float* __restrict__ out) {
    const int row = blockIdx.x;           // b*SEQ + s
    const int s   = row & (SEQ - 1);
    const int t   = threadIdx.x;          // covers hidden dims [4t, 4t+3]

    __shared__ v4f  s_row[256];           // staged embedding row (4KB)
    __shared__ float s_sum[8], s_sqs[8];
    __shared__ float s_mu, s_rstd;

    // Quantization params (uniform across grid)
    const float wmin  = keyToFloat(ws[0]);
    const float wmax  = keyToFloat(ws[1]);
    const float scale = (wmax - wmin) * (1.0f / 255.0f);   // (qmax-qmin)=255
    const float inv_s = 1.0f / scale;
    const float zp    = -128.0f - wmin * inv_s;

    const int tok = ids[row];

    // ---- async gather: each lane DMAs its 16B slice of weight[tok] into LDS
    unsigned lds_addr = (unsigned)(size_t)(&s_row[t]);     // low 32 bits = LDS offset
    unsigned long long gaddr =
        (unsigned long long)(size_t)(weight + (size_t)tok * HIDDEN + t * 4);
    asm volatile("global_load_async_to_lds_b128 %0, %1, off"
                 :: "v"(lds_addr), "v"(gaddr) : "memory");
    asm volatile("s_wait_asynccnt 0" ::: "memory");        // ASYNCcnt drain
    v4f w = s_row[t];                                      // own slot: no barrier needed

    // ---- dequantize (round-half-to-even matches jnp.round)
    v4f x;
    x.x = (fminf(fmaxf(rintf(w.x * inv_s + zp), -128.0f), 127.0f) - zp) * scale;
    x.y = (fminf(fmaxf(rintf(w.y * inv_s + zp), -128.0f), 127.0f) - zp) * scale;
    x.z = (fminf(fmaxf(rintf(w.z * inv_s + zp), -128.0f), 127.0f) - zp) * scale;
    x.w = (fminf(fmaxf(rintf(w.w * inv_s + zp), -128.0f), 127.0f) - zp) * scale;

    // ---- positional encoding: pe[s, 2i]=sin(s*f_i), pe[s, 2i+1]=cos(s*f_i),
    //      f_i = exp(2i * -ln(10000)/HIDDEN)
    const float kC  = -9.210340371976184f / (float)HIDDEN; // -ln(10000)/H
    const int   d0  = t * 4;
    const float pos = (float)s;
    const float f0  = __expf((float)d0 * kC);
    const float f2  = __expf((float)(d0 + 2) * kC);
    float sn0, cs0, sn2, cs2;
    __sincosf(pos * f0, &sn0, &cs0);
    __sincosf(pos * f2, &sn2, &cs2);
    x.x += sn0; x.y += cs0; x.z += sn2; x.w += cs2;

    // ---- LayerNorm stats: block reduce sum & sum-of-squares over 1024
    float sum = x.x + x.y + x.z + x.w;
    float sqs = x.x * x.x + x.y * x.y + x.z * x.z + x.w * x.w;
    #pragma unroll
    for (int off = 16; off > 0; off >>= 1) {
        sum += __shfl_xor(sum, off, 32);
        sqs += __shfl_xor(sqs, off, 32);
    }
    const int wave = t >> 5;
    if ((t & 31) == 0) { s_sum[wave] = sum; s_sqs[wave] = sqs; }
    __syncthreads();
    if (t == 0) {
        float S = 0.0f, Q = 0.0f;
        #pragma unroll
        for (int i = 0; i < 8; ++i) { S += s_sum[i]; Q += s_sqs[i]; }
        const float mu  = S * (1.0f / HIDDEN);
        const float var = Q * (1.0f / HIDDEN) - mu * mu;
        s_mu   = mu;
        s_rstd = rsqrtf(var + EPS);
    }
    __syncthreads();
    const float mu = s_mu, r = s_rstd;

    // ---- scale/shift and store
    const v4f g = ((const v4f*)gamma)[t];
    const v4f b = ((const v4f*)beta)[t];
    v4f o;
    o.x = g.x * (x.x - mu) * r + b.x;
    o.y = g.y * (x.y - mu) * r + b.y;
    o.z = g.z * (x.z - mu) * r + b.z;
    o.w = g.w * (x.w - mu) * r + b.w;
    ((v4f*)out)[row * 256 + t] = o;
}

extern "C" void kernel_launch(void* const* d_in, const int* in_sizes, int n_in,
                              void* d_out, int out_size, void* d_ws, size_t ws_size,
                              hipStream_t stream) {
    const int*   ids    = (const int*)d_in[0];
    const float* weight = (const float*)d_in[1];
    const float* gamma  = (const float*)d_in[2];
    const float* beta   = (const float*)d_in[3];
    unsigned*    ws     = (unsigned*)d_ws;
    float*       out    = (float*)d_out;

    init_ws_kernel<<<1, 1, 0, stream>>>(ws);

    const int n4 = (VOCAB * HIDDEN) / 4;            // 12,865,792 float4s
    minmax_kernel<<<2048, 256, 0, stream>>>(weight, ws, n4);

    embed_ln_kernel<<<BATCH * SEQ, 256, 0, stream>>>(ids, weight, gamma, beta, ws, out);
}